// MAEInterface_12154757448402
// MI455X (gfx1250) — compile-verified
//
#include <hip/hip_runtime.h>
#include <math.h>

#define BDIM 64
#define ADIM 2048
#define CDIM 512
#define HDIM 1024
#define SDIM 3
#define PDIM (2 * CDIM + 1 + SDIM)   // 1028

typedef float v2f __attribute__((ext_vector_type(2)));
typedef float v8f __attribute__((ext_vector_type(8)));

// ---------------------------------------------------------------------------
// Kernel 1: params = ctrl_hidden @ W^T + b  via V_WMMA_F32_16X16X4_F32
// One wave (32 threads) per 16x16 output tile. M=64 -> 4 tiles, N=1028 -> 65
// tiles (tail masked). K = 1024 -> 256 wmma steps, fp32-exact accumulate.
// ---------------------------------------------------------------------------
__global__ void __launch_bounds__(32)
gemm_params_wmma(const float* __restrict__ ctrl,   // [64][1024]
                 const float* __restrict__ W,      // [1028][1024]
                 const float* __restrict__ bias,   // [1028]
                 float* __restrict__ params)       // [64][1028]
{
    const int tile = blockIdx.x;          // 0..259
    const int mt   = tile & 3;            // 4 M-tiles
    const int nt   = tile >> 2;           // 65 N-tiles
    const int lane = threadIdx.x;         // 0..31, EXEC all ones
    const int m    = lane & 15;           // A row / C-D column index
    const int h    = lane >> 4;           // lane half
    const int n    = lane & 15;           // B column
    const int m0   = mt * 16;
    const int p0   = nt * 16;
    const int p    = p0 + n;

    const bool  pvalid = (p < PDIM);
    const float pmask  = pvalid ? 1.0f : 0.0f;

    const float* arow = ctrl + (size_t)(m0 + m) * HDIM;
    const float* brow = W + (size_t)(pvalid ? p : 0) * HDIM;   // clamped, masked below

    v8f c = {};
#pragma unroll 4
    for (int k0 = 0; k0 < HDIM; k0 += 4) {
        const int kk = k0 + 2 * h;        // per ISA: VGPR j, lane-half h -> K = j + 2h
        v2f a, b2;
        a[0]  = arow[kk + 0];
        a[1]  = arow[kk + 1];
        b2[0] = brow[kk + 0] * pmask;
        b2[1] = brow[kk + 1] * pmask;
        c = __builtin_amdgcn_wmma_f32_16x16x4_f32(
                /*neg_a=*/false, a, /*neg_b=*/false, b2,
                /*c_mod=*/(short)0, c, /*reuse_a=*/false, /*reuse_b=*/false);
    }

    if (pvalid) {
        const float bb = bias[p];
#pragma unroll
        for (int r = 0; r < 8; ++r) {
            const int row = m0 + r + 8 * h;          // C/D: VGPR r, half h -> M = r + 8h
            params[(size_t)row * PDIM + p] = c[r] + bb;
        }
    }
}

// ---------------------------------------------------------------------------
// Kernel 2: per-batch — erase/add sigmoids, shift softmax, softplus gamma,
// circular conv + pow + L1 normalize -> write_att, shift output.
// One block (256 threads) per batch; 8 attention elements per thread.
// ---------------------------------------------------------------------------
__global__ void __launch_bounds__(256)
attention_kernel(const float* __restrict__ params,     // [64][1028]
                 const float* __restrict__ prev_att,   // [64][2048]
                 float* __restrict__ erase_ws,         // [64][512]
                 float* __restrict__ add_ws,           // [64][512]
                 float* __restrict__ write_att,        // [64][2048] (d_out slice)
                 float* __restrict__ shift_out)        // [64][3]    (d_out slice)
{
    const int b = blockIdx.x;
    const int t = threadIdx.x;
    const float* pr = params + (size_t)b * PDIM;

    // erase / add sigmoids (512 each, 2 per thread)
    for (int cc = t; cc < CDIM; cc += 256) {
        const float e = pr[SDIM + 1 + cc];
        const float a = pr[SDIM + 1 + CDIM + cc];
        erase_ws[b * CDIM + cc] = 1.0f / (1.0f + expf(-e));
        add_ws[b * CDIM + cc]   = 1.0f / (1.0f + expf(-a));
    }

    // shift softmax + gamma (redundantly in every lane; 4 cached scalar loads)
    const float s0 = pr[0], s1 = pr[1], s2 = pr[2], g = pr[3];
    const float mx  = fmaxf(s0, fmaxf(s1, s2));
    const float e0  = expf(s0 - mx), e1 = expf(s1 - mx), e2 = expf(s2 - mx);
    const float inv = 1.0f / (e0 + e1 + e2);
    const float sh0 = e0 * inv, sh1 = e1 * inv, sh2 = e2 * inv;
    // stable softplus(g) + 1
    const float gamma = fmaxf(g, 0.0f) + log1pf(expf(-fabsf(g))) + 1.0f;

    const float* pa = prev_att + (size_t)b * ADIM;
    float pw[8];
    float local = 0.0f;
#pragma unroll
    for (int i = 0; i < 8; ++i) {
        const int a   = t + i * 256;
        const int am1 = (a - 1) & (ADIM - 1);
        const int ap1 = (a + 1) & (ADIM - 1);
        const float shifted = pa[am1] * sh0 + pa[a] * sh1 + pa[ap1] * sh2;
        const float pv = powf(shifted, gamma);
        pw[i]  = pv;
        local += fabsf(pv);
    }

    __shared__ float red[256];
    red[t] = local;
    __syncthreads();
    for (int off = 128; off > 0; off >>= 1) {
        if (t < off) red[t] += red[t + off];
        __syncthreads();
    }
    const float invd = 1.0f / fmaxf(red[0], 1e-12f);

#pragma unroll
    for (int i = 0; i < 8; ++i)
        write_att[(size_t)b * ADIM + t + i * 256] = pw[i] * invd;

    if (t < 3)
        shift_out[b * 3 + t] = (t == 0) ? sh0 : ((t == 1) ? sh1 : sh2);
}

// ---------------------------------------------------------------------------
// Kernel 3: memory = prev_mem * (1 - w*erase) + w*add   (bandwidth phase)
// float4 per thread: 16.7M threads, pure B128 streaming (~512 MB traffic).
// ---------------------------------------------------------------------------
__global__ void __launch_bounds__(256)
memory_update(const float4* __restrict__ prev_mem,   // [64*2048*128] float4
              const float*  __restrict__ write_att,  // [64*2048]
              const float*  __restrict__ erase_ws,   // [64][512]
              const float*  __restrict__ add_ws,     // [64][512]
              float4* __restrict__ mem_out)
{
    const size_t idx4 = (size_t)blockIdx.x * blockDim.x + threadIdx.x; // 0..2^24-1
    const int    c4   = (int)(idx4 & 127);     // C/4 = 128
    const size_t ba   = idx4 >> 7;             // b*A + a
    const int    b    = (int)(ba >> 11);       // A = 2048

    const float w = write_att[ba];
    const float4 m = prev_mem[idx4];
    const float4 e = ((const float4*)(erase_ws + b * CDIM))[c4];
    const float4 a = ((const float4*)(add_ws   + b * CDIM))[c4];

    float4 o;
    o.x = m.x * (1.0f - w * e.x) + w * a.x;
    o.y = m.y * (1.0f - w * e.y) + w * a.y;
    o.z = m.z * (1.0f - w * e.z) + w * a.z;
    o.w = m.w * (1.0f - w * e.w) + w * a.w;
    mem_out[idx4] = o;
}

// ---------------------------------------------------------------------------
extern "C" void kernel_launch(void* const* d_in, const int* in_sizes, int n_in,
                              void* d_out, int out_size, void* d_ws, size_t ws_size,
                              hipStream_t stream)
{
    const float* ctrl     = (const float*)d_in[0];  // [64][1024]
    const float* prev_mem = (const float*)d_in[1];  // [64][2048][512]
    const float* prev_att = (const float*)d_in[2];  // [64][2048][1]
    const float* W        = (const float*)d_in[3];  // [1028][1024]
    const float* bias     = (const float*)d_in[4];  // [1028]

    float* out       = (float*)d_out;
    float* mem_out   = out;                                   // 64*2048*512
    float* write_att = out + (size_t)BDIM * ADIM * CDIM;      // 64*2048
    float* shift_out = write_att + (size_t)BDIM * ADIM;       // 64*3

    float* ws       = (float*)d_ws;
    float* params   = ws;                                     // 64*1028
    float* erase_ws = params + (size_t)BDIM * PDIM;           // 64*512
    float* add_ws   = erase_ws + (size_t)BDIM * CDIM;         // 64*512

    // 1) params GEMM on the matrix pipe (fp32-exact WMMA)
    gemm_params_wmma<<<4 * ((PDIM + 15) / 16), 32, 0, stream>>>(ctrl, W, bias, params);

    // 2) per-batch nonlinearities + attention shift/pow/normalize
    attention_kernel<<<BDIM, 256, 0, stream>>>(params, prev_att, erase_ws, add_ws,
                                               write_att, shift_out);

    // 3) streaming memory update (bandwidth-bound phase)
    const size_t n4 = (size_t)BDIM * ADIM * CDIM / 4;         // 16,777,216
    memory_update<<<(unsigned)(n4 / 256), 256, 0, stream>>>((const float4*)prev_mem,
                                                            write_att, erase_ws, add_ws,
                                                            (float4*)mem_out);
}